// TNEModel_18391049961805
// MI455X (gfx1250) — compile-verified
//
#include <hip/hip_runtime.h>

typedef float v2f __attribute__((ext_vector_type(2)));
typedef float v4f __attribute__((ext_vector_type(4)));
typedef float v8f __attribute__((ext_vector_type(8)));

#define Bn 8
#define Ln 4096
#define Dn 1024
#define Sn 64
#define Hn 1024
#define Fn 512
#define Cn 25

#define WMMA_F32(a, b, c) \
  __builtin_amdgcn_wmma_f32_16x16x4_f32(false, (a), false, (b), (short)0, (c), false, false)

// ---------------------------------------------------------------------------
// Kernel 1: span gather: span_emb[b*S+s, :] = emb[b, s1, :] - emb[b, s0, :]
// 256 threads x float4 = one 1024-float row per block.
// ---------------------------------------------------------------------------
__global__ void span_gather_kernel(const float* __restrict__ emb,
                                   const int* __restrict__ spans,
                                   float* __restrict__ span_emb) {
  const int row = blockIdx.x;             // b*S + s
  const int b  = row / Sn;
  const int s0 = spans[row * 2 + 0];
  const int s1 = spans[row * 2 + 1];
  const v4f* e0 = (const v4f*)(emb + ((size_t)b * Ln + s0) * Dn);
  const v4f* e1 = (const v4f*)(emb + ((size_t)b * Ln + s1) * Dn);
  v4f* o = (v4f*)(span_emb + (size_t)row * Dn);
  const int d = threadIdx.x;              // 256 threads, Dn/4 = 256 vec4
  o[d] = e1[d] - e0[d];
}

// ---------------------------------------------------------------------------
// Kernel 2: fp32 WMMA GEMM, 32x32 output per wave: Out = act(A[M,K]@W[K,N]+b)
// Per k-step: 2 A-frags + 2 B-frags -> 4 independent V_WMMA_F32_16X16X4_F32.
// A-frag (16x4 fp32): lanes 0-15 rows M, K={k,k+1}; lanes 16-31 K={k+2,k+3}.
// B-frag (4x16 fp32): lanes 0-15 cols N rows K={k,k+1}; lanes 16-31 K={k+2,k+3}.
// ---------------------------------------------------------------------------
__global__ void gemm_f32_wmma(const float* __restrict__ A, int lda,
                              const float* __restrict__ W, int ldw,
                              const float* __restrict__ bias,
                              float* __restrict__ Out, int ldo,
                              int K, int relu) {
  const int lane = threadIdx.x & 31;
  const int l15  = lane & 15;
  const int kh   = (lane >> 4) * 2;       // 0 or 2
  const int n0   = blockIdx.x * 32;
  const int m0   = blockIdx.y * 32;

  const float* arow0 = A + (size_t)(m0 + l15) * lda;
  const float* arow1 = arow0 + (size_t)16 * lda;

  v8f acc00 = {}, acc01 = {}, acc10 = {}, acc11 = {};
#pragma unroll 2
  for (int k = 0; k < K; k += 4) {
    const int kk = k + kh;
    v2f a0 = *(const v2f*)(arow0 + kk);
    v2f a1 = *(const v2f*)(arow1 + kk);
    const float* wr = W + (size_t)kk * ldw + n0 + l15;
    v2f b0, b1;
    b0.x = wr[0];   b0.y = wr[ldw];
    b1.x = wr[16];  b1.y = wr[ldw + 16];
    acc00 = WMMA_F32(a0, b0, acc00);
    acc01 = WMMA_F32(a0, b1, acc01);
    acc10 = WMMA_F32(a1, b0, acc10);
    acc11 = WMMA_F32(a1, b1, acc11);
  }

  // C/D layout: VGPR r holds (M=r, lanes 0-15) / (M=r+8, lanes 16-31), N=lane%16
  const int mh = (lane >> 4) * 8;
  const int nc0 = n0 + l15;
  const int nc1 = nc0 + 16;
  const float bv0 = bias ? bias[nc0] : 0.0f;
  const float bv1 = bias ? bias[nc1] : 0.0f;
#pragma unroll
  for (int r = 0; r < 8; ++r) {
    const size_t row0 = (size_t)(m0 + r + mh) * ldo;
    const size_t row1 = (size_t)(m0 + 16 + r + mh) * ldo;
    float v00 = acc00[r] + bv0, v01 = acc01[r] + bv1;
    float v10 = acc10[r] + bv0, v11 = acc11[r] + bv1;
    if (relu) {
      v00 = fmaxf(v00, 0.0f); v01 = fmaxf(v01, 0.0f);
      v10 = fmaxf(v10, 0.0f); v11 = fmaxf(v11, 0.0f);
    }
    Out[row0 + nc0] = v00;  Out[row0 + nc1] = v01;
    Out[row1 + nc0] = v10;  Out[row1 + nc1] = v11;
  }
}

// ---------------------------------------------------------------------------
// Kernel 3: fused pair classification, one (b,i) per block, 2 waves.
// hidden[j,k] = relu(Ap[b,j,k] + Cp[b,i,k] + pb1[k])   (M=64 j's, K=1024)
// logits[b,i,j,:] = hidden[j,:] @ pW2[H,25] + pb2       (N=25 padded to 32)
// Wave w handles j in [w*32, w*32+32): 2x2 grid of 16x16 WMMA tiles.
// ---------------------------------------------------------------------------
__global__ void pair_logits_kernel(const float* __restrict__ Ap,   // [B*S, H]
                                   const float* __restrict__ Cp,   // [B*S, H]
                                   const float* __restrict__ pb1,  // [H]
                                   const float* __restrict__ pW2,  // [H, C]
                                   const float* __restrict__ pb2,  // [C]
                                   float* __restrict__ out) {      // [B*S*S, C]
  __shared__ float cvec[Hn];
  const int bi = blockIdx.x;              // b*S + i
  const int b  = bi / Sn;
  const float* crow = Cp + (size_t)bi * Hn;
  for (int k = threadIdx.x; k < Hn; k += blockDim.x)
    cvec[k] = crow[k] + pb1[k];
  __syncthreads();

  const int wave = threadIdx.x >> 5;      // 0..1
  const int lane = threadIdx.x & 31;
  const int l15  = lane & 15;
  const int kh   = (lane >> 4) * 2;
  const int j0   = wave * 32;
  const bool n1ok = (16 + l15) < Cn;      // high class tile: cols 16..24 valid

  const float* arow0 = Ap + ((size_t)(b * Sn) + j0 + l15) * Hn;
  const float* arow1 = arow0 + (size_t)16 * Hn;

  v8f acc00 = {}, acc01 = {}, acc10 = {}, acc11 = {};
#pragma unroll 2
  for (int k = 0; k < Hn; k += 4) {
    const int kk = k + kh;
    v2f c  = *(const v2f*)(cvec + kk);            // ds_load_b64
    v2f a0 = *(const v2f*)(arow0 + kk);           // global_load_b64
    v2f a1 = *(const v2f*)(arow1 + kk);
    a0.x = fmaxf(a0.x + c.x, 0.0f);  a0.y = fmaxf(a0.y + c.y, 0.0f);
    a1.x = fmaxf(a1.x + c.x, 0.0f);  a1.y = fmaxf(a1.y + c.y, 0.0f);
    const float* wr = pW2 + (size_t)kk * Cn;
    v2f b0, b1;
    b0.x = wr[l15];                        b0.y = wr[Cn + l15];
    b1.x = n1ok ? wr[16 + l15]      : 0.0f;
    b1.y = n1ok ? wr[Cn + 16 + l15] : 0.0f;
    acc00 = WMMA_F32(a0, b0, acc00);
    acc01 = WMMA_F32(a0, b1, acc01);
    acc10 = WMMA_F32(a1, b0, acc10);
    acc11 = WMMA_F32(a1, b1, acc11);
  }

  const int mh  = (lane >> 4) * 8;
  const float bv0 = pb2[l15];                         // l15 < 16 < 25: valid
  const float bv1 = n1ok ? pb2[16 + l15] : 0.0f;
#pragma unroll
  for (int r = 0; r < 8; ++r) {
    const int ja = j0 + r + mh;
    const int jb = ja + 16;
    const size_t rowA = ((size_t)bi * Sn + ja) * Cn;
    const size_t rowB = ((size_t)bi * Sn + jb) * Cn;
    out[rowA + l15] = acc00[r] + bv0;
    out[rowB + l15] = acc10[r] + bv0;
    if (n1ok) {
      out[rowA + 16 + l15] = acc01[r] + bv1;
      out[rowB + 16 + l15] = acc11[r] + bv1;
    }
  }
}

// ---------------------------------------------------------------------------
extern "C" void kernel_launch(void* const* d_in, const int* in_sizes, int n_in,
                              void* d_out, int out_size, void* d_ws, size_t ws_size,
                              hipStream_t stream) {
  const float* emb  = (const float*)d_in[0];
  const int*   spn  = (const int*)  d_in[1];
  const float* aW1  = (const float*)d_in[2];
  const float* ab1  = (const float*)d_in[3];
  const float* aW2  = (const float*)d_in[4];
  const float* ab2  = (const float*)d_in[5];
  const float* cW1  = (const float*)d_in[6];
  const float* cb1  = (const float*)d_in[7];
  const float* cW2  = (const float*)d_in[8];
  const float* cb2  = (const float*)d_in[9];
  const float* pW1  = (const float*)d_in[10];
  const float* pb1  = (const float*)d_in[11];
  const float* pW2  = (const float*)d_in[12];
  const float* pb2  = (const float*)d_in[13];
  float* out = (float*)d_out;

  const int R = Bn * Sn;                  // 512 span rows
  float* ws   = (float*)d_ws;
  float* span = ws;                       // [512,1024]
  float* Ha   = span + (size_t)R * Dn;    // [512,1024]
  float* Hc   = Ha   + (size_t)R * Hn;    // [512,1024]
  float* Ar   = Hc   + (size_t)R * Hn;    // [512,512]
  float* Cr   = Ar   + (size_t)R * Fn;    // [512,512]
  float* Ap   = Cr   + (size_t)R * Fn;    // [512,1024]
  float* Cp   = Ap   + (size_t)R * Hn;    // [512,1024]
  (void)in_sizes; (void)n_in; (void)out_size; (void)ws_size;

  // 1) span embeddings
  span_gather_kernel<<<R, 256, 0, stream>>>(emb, spn, span);

  const dim3 blk(32, 1, 1);
  // 2) anchor / complement MLPs
  gemm_f32_wmma<<<dim3(Hn / 32, R / 32), blk, 0, stream>>>(span, Dn, aW1, Hn, ab1, Ha, Hn, Dn, 1);
  gemm_f32_wmma<<<dim3(Hn / 32, R / 32), blk, 0, stream>>>(span, Dn, cW1, Hn, cb1, Hc, Hn, Dn, 1);
  gemm_f32_wmma<<<dim3(Fn / 32, R / 32), blk, 0, stream>>>(Ha, Hn, aW2, Fn, ab2, Ar, Fn, Hn, 0);
  gemm_f32_wmma<<<dim3(Fn / 32, R / 32), blk, 0, stream>>>(Hc, Hn, cW2, Fn, cb2, Cr, Fn, Hn, 0);

  // 3) split pW1: A' = anchor @ pW1[:F], C' = complement @ pW1[F:]
  gemm_f32_wmma<<<dim3(Hn / 32, R / 32), blk, 0, stream>>>(Ar, Fn, pW1,                  Hn, nullptr, Ap, Hn, Fn, 0);
  gemm_f32_wmma<<<dim3(Hn / 32, R / 32), blk, 0, stream>>>(Cr, Fn, pW1 + (size_t)Fn*Hn,  Hn, nullptr, Cp, Hn, Fn, 0);

  // 4) fused relu(A'[j]+C'[i]+pb1) @ pW2 + pb2 over all S^2 pairs per batch
  pair_logits_kernel<<<R, 64, 0, stream>>>(Ap, Cp, pb1, pW2, pb2, out);
}